// GaussRenderer_44839458570277
// MI455X (gfx1250) — compile-verified
//
#include <hip/hip_runtime.h>
#include <cstdint>

typedef __attribute__((ext_vector_type(2))) float    v2f;
typedef __attribute__((ext_vector_type(8))) float    v8f;
typedef __attribute__((ext_vector_type(4))) unsigned u32x4;
typedef __attribute__((ext_vector_type(8))) int      i32x8;
typedef __attribute__((ext_vector_type(4))) int      i32x4;

#define NG    2048
#define IMG_W 256
#define IMG_H 256
#define REC   16   // floats per gaussian record

// record layout: 0:m2x 1:m2y 2:conicA 3:conicB 4:conicC 5:opac 6:r 7:g 8:b 9:depth
//               10:rminx 11:rminy 12:rmaxx 13:rmaxy 14:const 1.0 15:const 0.0

// ---------------------------------------------------------------------------
// Kernel 1: per-gaussian preprocess
// ---------------------------------------------------------------------------
__global__ __launch_bounds__(256) void gs_prep(
    const float* __restrict__ means, const float* __restrict__ scales,
    const float* __restrict__ rots,  const float* __restrict__ shs,
    const float* __restrict__ opac,  const float* __restrict__ V,
    const float* __restrict__ P,     const float* __restrict__ cam,
    float* __restrict__ rec, float* __restrict__ keys) {
  int i = blockIdx.x * blockDim.x + threadIdx.x;
  if (i >= NG) return;

  const float TANF = 0.57735026918962576f;      // tan(30 deg)
  const float FX   = 221.70250336881628f;       // 256 / (2*TAN)
  const float FY   = 221.70250336881628f;

  float mx = means[3*i+0], my = means[3*i+1], mz = means[3*i+2];

  // p_view = [m,1] @ V   (row-vector convention)
  float pv0 = mx*V[0] + my*V[4] + mz*V[8]  + V[12];
  float pv1 = mx*V[1] + my*V[5] + mz*V[9]  + V[13];
  float pv2 = mx*V[2] + my*V[6] + mz*V[10] + V[14];
  float pv3 = mx*V[3] + my*V[7] + mz*V[11] + V[15];

  // hom = p_view @ P
  float h0 = pv0*P[0] + pv1*P[4] + pv2*P[8]  + pv3*P[12];
  float h1 = pv0*P[1] + pv1*P[5] + pv2*P[9]  + pv3*P[13];
  float h3 = pv0*P[3] + pv1*P[7] + pv2*P[11] + pv3*P[15];
  float iw  = 1.0f / (h3 + 1e-6f);
  float m2x = ((h0*iw + 1.0f)*(float)IMG_W - 1.0f)*0.5f;
  float m2y = ((h1*iw + 1.0f)*(float)IMG_H - 1.0f)*0.5f;
  float depth   = pv2;
  float infront = (pv2 >= 0.2f) ? 1.0f : 0.0f;

  // SH degree-3 color (dirs NOT normalized, per reference)
  float dx = mx - cam[0], dy = my - cam[1], dz = mz - cam[2];
  float xx=dx*dx, yy=dy*dy, zz=dz*dz, xy=dx*dy, yz=dy*dz, xz=dx*dz;
  float b[16];
  b[0]  = 0.28209479177387814f;
  b[1]  = -0.4886025119029199f*dy;
  b[2]  =  0.4886025119029199f*dz;
  b[3]  = -0.4886025119029199f*dx;
  b[4]  =  1.0925484305920792f*xy;
  b[5]  = -1.0925484305920792f*yz;
  b[6]  =  0.31539156525252005f*(2.f*zz - xx - yy);
  b[7]  = -1.0925484305920792f*xz;
  b[8]  =  0.5462742152960396f*(xx - yy);
  b[9]  = -0.5900435899266435f*dy*(3.f*xx - yy);
  b[10] =  2.890611442640554f*xy*dz;
  b[11] = -0.4570457994644658f*dy*(4.f*zz - xx - yy);
  b[12] =  0.3731763325901154f*dz*(2.f*zz - 3.f*xx - 3.f*yy);
  b[13] = -0.4570457994644658f*dx*(4.f*zz - xx - yy);
  b[14] =  1.445305721320277f*dz*(xx - yy);
  b[15] = -0.5900435899266435f*dx*(xx - 3.f*yy);
  float col[3];
  #pragma unroll
  for (int c = 0; c < 3; ++c) {
    float r = 0.f;
    #pragma unroll
    for (int k = 0; k < 16; ++k) r += b[k]*shs[i*48 + k*3 + c];
    col[c] = fmaxf(r + 0.5f, 0.f);
  }

  // quaternion -> rotation
  float q0=rots[4*i],q1=rots[4*i+1],q2=rots[4*i+2],q3=rots[4*i+3];
  float qn = rsqrtf(q0*q0+q1*q1+q2*q2+q3*q3);
  float qr=q0*qn, qx=q1*qn, qy=q2*qn, qz=q3*qn;
  float R[3][3];
  R[0][0]=1.f-2.f*(qy*qy+qz*qz); R[0][1]=2.f*(qx*qy-qr*qz); R[0][2]=2.f*(qx*qz+qr*qy);
  R[1][0]=2.f*(qx*qy+qr*qz); R[1][1]=1.f-2.f*(qx*qx+qz*qz); R[1][2]=2.f*(qy*qz-qr*qx);
  R[2][0]=2.f*(qx*qz-qr*qy); R[2][1]=2.f*(qy*qz+qr*qx); R[2][2]=1.f-2.f*(qx*qx+qy*qy);
  float s2v[3] = { scales[3*i]*scales[3*i], scales[3*i+1]*scales[3*i+1],
                   scales[3*i+2]*scales[3*i+2] };
  float S[3][3];
  #pragma unroll
  for (int a = 0; a < 3; ++a)
    #pragma unroll
    for (int d = 0; d < 3; ++d)
      S[a][d] = R[a][0]*s2v[0]*R[d][0] + R[a][1]*s2v[1]*R[d][1] + R[a][2]*s2v[2]*R[d][2];

  // cov2d = J * Wm * S * Wm^T * J^T  (Wm = V3x3^T); fold T = J*Wm (rows 0,1)
  float tz = pv2, invz = 1.f/tz;
  float lim = 1.3f*TANF;
  float txl = fminf(fmaxf(pv0*invz,-lim),lim)*tz;
  float tyl = fminf(fmaxf(pv1*invz,-lim),lim)*tz;
  float J00 = FX*invz, J02 = -txl*invz*invz*FX;
  float J11 = FY*invz, J12 = -tyl*invz*invz*FY;
  float T0[3], T1[3];
  #pragma unroll
  for (int bb = 0; bb < 3; ++bb) {   // Wm[a][b] = V[b*4 + a]
    T0[bb] = J00*V[bb*4+0] + J02*V[bb*4+2];
    T1[bb] = J11*V[bb*4+1] + J12*V[bb*4+2];
  }
  float u[3], w[3];
  #pragma unroll
  for (int bb = 0; bb < 3; ++bb) {
    u[bb] = S[bb][0]*T0[0] + S[bb][1]*T0[1] + S[bb][2]*T0[2];
    w[bb] = S[bb][0]*T1[0] + S[bb][1]*T1[1] + S[bb][2]*T1[2];
  }
  float c00 = T0[0]*u[0]+T0[1]*u[1]+T0[2]*u[2] + 0.3f;
  float c01 = T1[0]*u[0]+T1[1]*u[1]+T1[2]*u[2];
  float c11 = T1[0]*w[0]+T1[1]*w[1]+T1[2]*w[2] + 0.3f;

  float det = c00*c11 - c01*c01;
  float mid = 0.5f*(c00 + c11);
  float sq  = sqrtf(fmaxf(mid*mid - det, 0.1f));
  float lam = fmaxf(mid + sq, mid - sq);
  float radii = 3.f*ceilf(sqrtf(lam));
  float rminx = fminf(fmaxf(m2x - radii, 0.f), 255.f);
  float rminy = fminf(fmaxf(m2y - radii, 0.f), 255.f);
  float rmaxx = fminf(fmaxf(m2x + radii, 0.f), 255.f);
  float rmaxy = fminf(fmaxf(m2y + radii, 0.f), 255.f);
  float idet = 1.f/det;

  float* r = rec + i*REC;
  r[0]=m2x; r[1]=m2y; r[2]=c11*idet; r[3]=c00*idet; r[4]=-2.f*c01*idet;
  r[5]=opac[i]*infront;
  r[6]=col[0]; r[7]=col[1]; r[8]=col[2]; r[9]=depth;
  r[10]=rminx; r[11]=rminy; r[12]=rmaxx; r[13]=rmaxy;
  r[14]=1.f;   // constant 1.0 -> 'acc' channel column of B
  r[15]=0.f;   // constant 0.0 -> unused B columns
  keys[i]=depth;
}

// ---------------------------------------------------------------------------
// Kernel 2: single-workgroup bitonic argsort of 2048 depths (ascending)
// ---------------------------------------------------------------------------
__global__ __launch_bounds__(1024) void gs_sort(const float* __restrict__ keys,
                                                int* __restrict__ order) {
  __shared__ float sk[NG];
  __shared__ int   sv[NG];
  int t = threadIdx.x;
  for (int i = t; i < NG; i += 1024) { sk[i] = keys[i]; sv[i] = i; }
  for (int size = 2; size <= NG; size <<= 1) {
    for (int stride = size >> 1; stride > 0; stride >>= 1) {
      __syncthreads();
      int pos     = 2*t - (t & (stride - 1));
      int partner = pos + stride;
      bool up = ((pos & size) == 0);
      float k0 = sk[pos], k1 = sk[partner];
      bool swp = up ? (k0 > k1) : (k0 < k1);
      if (swp) {
        sk[pos] = k1; sk[partner] = k0;
        int v0 = sv[pos]; sv[pos] = sv[partner]; sv[partner] = v0;
      }
    }
  }
  __syncthreads();
  for (int i = t; i < NG; i += 1024) order[i] = sv[i];
}

// ---------------------------------------------------------------------------
// Kernel 3: gather records into depth order (AoS, 64B/record for TDM)
// ---------------------------------------------------------------------------
__global__ __launch_bounds__(256) void gs_gather(const float* __restrict__ rec,
                                                 const int* __restrict__ order,
                                                 float* __restrict__ srec) {
  int idx = blockIdx.x*256 + threadIdx.x;
  if (idx < NG*REC) {
    int g = idx >> 4, f = idx & 15;
    srec[idx] = rec[order[g]*REC + f];
  }
}

// half-wave swap (lane ^ 16) without touching the DS pipe if possible
__device__ __forceinline__ float half_swap(float x) {
#if __has_builtin(__builtin_amdgcn_permlanex16)
  int r = __builtin_amdgcn_permlanex16(__float_as_int(x), __float_as_int(x),
                                       0x76543210, 0xfedcba98, false, false);
  return __int_as_float(r);
#else
  return __shfl_xor(x, 16, 32);
#endif
}

// ---------------------------------------------------------------------------
// Kernel 4: tiled alpha-blend render.
//   grid = 16 tiles x 32 row-blocks (512 blocks); block = 256 threads = 8 waves.
//   Whole sorted splat table (128KB) staged to LDS with one TDM descriptor.
//   Each wave owns one 16-pixel group: 16 pixels x 4 gaussians per step ->
//   V_WMMA_F32_16X16X4_F32, accumulator columns = {r,g,b,depth,acc}.
// ---------------------------------------------------------------------------
__global__ __launch_bounds__(256) void gs_render(const float* __restrict__ srec,
                                                 float* __restrict__ out) {
  extern __shared__ float lds[];

  // --- TDM: DMA 2048*16 floats (as 16384 x 8B elements, 1 row) into LDS ---
  if (threadIdx.x < 32) {
    unsigned ldsOff = __builtin_amdgcn_groupstaticsize();
    unsigned long long ga = (unsigned long long)(uintptr_t)srec;
    u32x4 g0 = { 1u,                                    // count=1 (valid), no gather
                 ldsOff,                                // lds_addr
                 (unsigned)ga,                          // global_addr[31:0]
                 (unsigned)((ga >> 32) & 0x1FFFFFFull) | (2u << 30) }; // addr[56:32] | type=2
    i32x8 g1 = { (int)0x30000,            // data_size=3 (8B), mask/flags=0
                 (int)(16384u << 16),     // tensor_dim0[15:0]=16384 (<<16), barrier addr=0
                 (int)(1u << 16),         // tensor_dim0[31:16]=0, tensor_dim1=1
                 (int)(16384u << 16),     // tile_dim0=16384
                 (int)1,                  // tile_dim1=1, tile_dim2=0
                 (int)16384,              // tensor_dim0_stride
                 0, 0 };
    i32x4 gz4 = {0,0,0,0};
    i32x8 gz8 = {0,0,0,0,0,0,0,0};
    __builtin_amdgcn_tensor_load_to_lds(g0, g1, gz4, gz4, gz8, 0);
    __builtin_amdgcn_s_wait_tensorcnt(0);
  }
  __syncthreads();

  int tile = blockIdx.x >> 5;          // 16 tiles
  int rb   = blockIdx.x & 31;          // 32 row-blocks of 2 rows
  int tx0  = (tile & 3) * 64;
  int ty0  = (tile >> 2) * 64;

  // fold per-tile bbox mask into opacity field (done once per block)
  for (int g = threadIdx.x; g < NG; g += 256) {
    float* r = lds + g*REC;
    float tlx = fmaxf(r[10], (float)tx0),       tly = fmaxf(r[11], (float)ty0);
    float brx = fminf(r[12], (float)(tx0+63)),  bry = fminf(r[13], (float)(ty0+63));
    if (!((brx > tlx) && (bry > tly))) r[5] = 0.f;
  }
  __syncthreads();

  int lane = threadIdx.x & 31;
  int wv   = threadIdx.x >> 5;
  int n    = lane & 15;      // pixel row in A/D; channel column in B/D
  int h    = lane >> 4;      // half-wave: gaussian sub-pair {0,1} vs {2,3}

  // branchless channel offset for the B-matrix column this lane supplies:
  // n<4 -> {r,g,b,depth}, n==4 -> record[14]==1.0 (acc), n>4 -> record[15]==0.0
  int coff = (n < 4) ? (6 + n) : ((n == 4) ? 14 : 15);

  // one 16-pixel group per wave: 2 rows x 4 segments
  int row = wv >> 2, seg = wv & 3;
  int gy  = ty0 + rb*2 + row;
  int gx0 = tx0 + seg*16;
  float px = (float)(gx0 + n), py = (float)gy;

  v8f acc = {0.f,0.f,0.f,0.f,0.f,0.f,0.f,0.f};
  float T = 1.f;

  for (int base = 0; base < NG; base += 4) {
    const float* r0 = lds + (base + 2*h)*REC;   // this half's gaussian pair
    const float* r1 = r0 + REC;
    float dx0 = px - r0[0], dy0 = py - r0[1];
    float dx1 = px - r1[0], dy1 = py - r1[1];
    float e0 = -0.5f*(dx0*dx0*r0[2] + dy0*dy0*r0[3] + dx0*dy0*r0[4]);
    float e1 = -0.5f*(dx1*dx1*r1[2] + dy1*dy1*r1[3] + dx1*dy1*r1[4]);
    float a0 = fminf(__expf(e0)*r0[5], 0.99f);
    float a1 = fminf(__expf(e1)*r1[5], 0.99f);
    float om0 = 1.f - a0;
    float pp  = om0*(1.f - a1);                 // (1-a_even)(1-a_odd) of this half
    float other = half_swap(pp);                // partner half's product
    float Tpre = h ? (T*other) : T;             // ordered transmittance

    v2f A; A.x = Tpre*a0; A.y = Tpre*om0*a1;    // weights (A-matrix layout)
    v2f B; B.x = r0[coff]; B.y = r1[coff];      // channel column n (branchless)

    float p01 = h ? other : pp;                 // product of pair {0,1}
    float p23 = h ? pp : other;                 // product of pair {2,3}
    T = T * (p01 * p23);                        // identical sequence in both halves

    acc = __builtin_amdgcn_wmma_f32_16x16x4_f32(
        false, A, false, B, (short)0, acc, false, false);
  }

  // D layout: VGPR v holds pixel M=v (+8 for upper half), column N=n
  #pragma unroll
  for (int v = 0; v < 8; ++v) {
    float val = acc[v];
    float apx = __shfl(val, 4, 16);             // this pixel's acc (channel 4)
    int x = gx0 + v + 8*h;
    int p = gy*IMG_W + x;
    if (n < 3)       out[p*3 + n]            = val + (1.f - apx);
    else if (n == 3) out[IMG_W*IMG_H*3 + p]  = val;
    else if (n == 4) out[IMG_W*IMG_H*4 + p]  = val;
  }
}

// ---------------------------------------------------------------------------
extern "C" void kernel_launch(void* const* d_in, const int* in_sizes, int n_in,
                              void* d_out, int out_size, void* d_ws, size_t ws_size,
                              hipStream_t stream) {
  const float* means = (const float*)d_in[0];
  const float* scales= (const float*)d_in[1];
  const float* rots  = (const float*)d_in[2];
  const float* shs   = (const float*)d_in[3];
  const float* opac  = (const float*)d_in[4];
  const float* V     = (const float*)d_in[5];
  const float* P     = (const float*)d_in[6];
  const float* cam   = (const float*)d_in[7];
  float* out = (float*)d_out;

  float* rec   = (float*)d_ws;                  // NG*16 floats
  float* keys  = rec + NG*REC;                  // NG floats
  int*   order = (int*)(keys + NG);             // NG ints
  float* srec  = (float*)(order + NG);          // NG*16 floats (sorted, TDM src)

  gs_prep<<<NG/256, 256, 0, stream>>>(means, scales, rots, shs, opac, V, P, cam,
                                      rec, keys);
  gs_sort<<<1, 1024, 0, stream>>>(keys, order);
  gs_gather<<<(NG*REC)/256, 256, 0, stream>>>(rec, order, srec);

  (void)hipFuncSetAttribute((const void*)gs_render,
                            hipFuncAttributeMaxDynamicSharedMemorySize,
                            NG*REC*(int)sizeof(float));
  gs_render<<<512, 256, NG*REC*sizeof(float), stream>>>(srec, out);
}